// MultiheadFastAttentionBlock_32435593019952
// MI455X (gfx1250) — compile-verified
//
#include <hip/hip_runtime.h>
#include <math.h>

// Problem constants (fixed by setup_inputs): B=4, S=4096, D=256, M=256, H=8.
#define BB 4
#define SS 4096
#define DD 256
#define MM 256
#define HH 8

constexpr float LN_EPS = 1e-5f;
constexpr float KEPS   = 1e-4f;
constexpr float DNORM  = 0.25f;     // D^-0.25 = 256^-0.25
constexpr float RATIO  = 0.0625f;   // M^-0.5  = 256^-0.5
constexpr float DIAGC  = 0.03125f;  // 0.5 * DNORM^2

typedef __attribute__((ext_vector_type(2))) float v2f;
typedef __attribute__((ext_vector_type(8))) float v8f;

__device__ __forceinline__ v8f wmma4(v2f a, v2f b, v8f c) {
  // V_WMMA_F32_16X16X4_F32 : D = A(16x4) * B(4x16) + C(16x16), all fp32
  return __builtin_amdgcn_wmma_f32_16x16x4_f32(false, a, false, b, (short)0, c,
                                               false, false);
}

// ---------------- LayerNorm + diag ----------------
// One block per row (b,s); 256 threads == D elements.
__global__ void pf_ln_kernel(const float* __restrict__ x,
                             const float* __restrict__ gamma,
                             const float* __restrict__ beta,
                             float* __restrict__ xn,
                             float* __restrict__ diag) {
  __shared__ float sd[256];
  const int row = blockIdx.x;
  const int t   = threadIdx.x;
  // x is read exactly once in the whole pipeline: non-temporal (don't pollute L2).
  const float v = __builtin_nontemporal_load(&x[(size_t)row * DD + t]);

  sd[t] = v; __syncthreads();
  for (int s = 128; s > 0; s >>= 1) { if (t < s) sd[t] += sd[t + s]; __syncthreads(); }
  const float mu = sd[0] * (1.0f / DD); __syncthreads();

  const float c = v - mu;
  sd[t] = c * c; __syncthreads();
  for (int s = 128; s > 0; s >>= 1) { if (t < s) sd[t] += sd[t + s]; __syncthreads(); }
  const float var = sd[0] * (1.0f / DD); __syncthreads();

  const float y = c * rsqrtf(var + LN_EPS) * gamma[t] + beta[t];
  xn[(size_t)row * DD + t] = y;   // re-read by GEMM1/GEMM2: keep L2-resident (RT)

  sd[t] = y * y; __syncthreads();
  for (int s = 128; s > 0; s >>= 1) { if (t < s) sd[t] += sd[t + s]; __syncthreads(); }
  if (t == 0) diag[row] = DIAGC * sd[0];
}

// ---------------- GEMM1: P = (DNORM*xn) @ proj^T  [S,M] per batch -------------
// Per wave: 16(s) x 64(m) tile = 4 accumulators. 4096 waves -> 512 blocks.
__global__ void pf_gemm1_kernel(const float* __restrict__ xn,
                                const float* __restrict__ proj,
                                float* __restrict__ P) {
  const int warp = threadIdx.x >> 5;
  const int l    = threadIdx.x & 31;
  const int wg   = blockIdx.x * 8 + warp;        // [0, 4096)
  const int b    = wg >> 10;
  const int rem  = wg & 1023;
  const int s0   = (rem >> 2) << 4;              // 256 s-tiles
  const int m0   = (rem & 3) << 6;               // 4 m-groups of 64
  const int lc   = l & 15;                       // A row / B col
  const int kb   = (l >> 4) << 1;                // K sub-offset {0,2}

  const float* Abase = xn + ((size_t)(b * SS) + s0 + lc) * DD + kb;
  v8f a0 = {}, a1 = {}, a2 = {}, a3 = {};
  for (int k0 = 0; k0 < DD; k0 += 4) {
    v2f a = *(const v2f*)(Abase + k0);
    a *= DNORM;
    v2f b0 = *(const v2f*)(proj + (size_t)(m0 +  0 + lc) * DD + k0 + kb);
    v2f b1 = *(const v2f*)(proj + (size_t)(m0 + 16 + lc) * DD + k0 + kb);
    v2f b2 = *(const v2f*)(proj + (size_t)(m0 + 32 + lc) * DD + k0 + kb);
    v2f b3 = *(const v2f*)(proj + (size_t)(m0 + 48 + lc) * DD + k0 + kb);
    a0 = wmma4(a, b0, a0);
    a1 = wmma4(a, b1, a1);
    a2 = wmma4(a, b2, a2);
    a3 = wmma4(a, b3, a3);
  }
  const int rhalf = (l >> 4) << 3;
  for (int r = 0; r < 8; ++r) {
    size_t o = ((size_t)(b * SS) + s0 + r + rhalf) * MM + m0 + lc;
    P[o +  0] = a0[r];
    P[o + 16] = a1[r];
    P[o + 32] = a2[r];
    P[o + 48] = a3[r];
  }
}

// ---------------- q = ratio*(exp(P - diag - rowmax) + eps); save rowmax -------
__global__ void pf_qrow_kernel(const float* __restrict__ P,
                               const float* __restrict__ diag,
                               float* __restrict__ q,
                               float* __restrict__ rowmax) {
  __shared__ float sd[256];
  const int row = blockIdx.x;
  const int t   = threadIdx.x;                   // m index
  const float p = P[(size_t)row * MM + t];
  sd[t] = p; __syncthreads();
  for (int s = 128; s > 0; s >>= 1) { if (t < s) sd[t] = fmaxf(sd[t], sd[t + s]); __syncthreads(); }
  const float rm = sd[0];
  q[(size_t)row * MM + t] = RATIO * (expf(p - diag[row] - rm) + KEPS);
  if (t == 0) rowmax[row] = rm;
}

// ---------------- per-batch global max over (S,M) ----------------
__global__ void pf_bmax_kernel(const float* __restrict__ rowmax,
                               float* __restrict__ mx) {
  __shared__ float sd[256];
  const int b = blockIdx.x, t = threadIdx.x;
  float m = -3.0e38f;
  for (int s = t; s < SS; s += 256) m = fmaxf(m, rowmax[b * SS + s]);
  sd[t] = m; __syncthreads();
  for (int s = 128; s > 0; s >>= 1) { if (t < s) sd[t] = fmaxf(sd[t], sd[t + s]); __syncthreads(); }
  if (t == 0) mx[b] = sd[0];
}

__global__ void pf_zero_kernel(float* __restrict__ p, int n) {
  int i = blockIdx.x * 256 + threadIdx.x;
  if (i < n) p[i] = 0.0f;
}

// ---------------- k (in-place over P) + ksum column sums ----------------
// grid = B*16 blocks; thread = m; each block covers 256 s-rows of one batch.
__global__ void pf_kexp_kernel(float* __restrict__ P,
                               const float* __restrict__ diag,
                               const float* __restrict__ mx,
                               float* __restrict__ ksum) {
  const int b     = blockIdx.x >> 4;
  const int chunk = blockIdx.x & 15;
  const int t     = threadIdx.x;                 // m index
  const float mxb = mx[b];
  float local = 0.0f;
  const int sBeg = chunk * (SS / 16);
  for (int s = sBeg; s < sBeg + SS / 16; ++s) {
    const size_t idx = ((size_t)b * SS + s) * MM + t;
    const float kv = RATIO * (expf(P[idx] - diag[b * SS + s] - mxb) + KEPS);
    P[idx] = kv;
    local += kv;
  }
  atomicAdd(&ksum[b * MM + t], local);
}

// ---------------- GEMM2: ctx[m,d] += sum_s k[s,m]*xn[s,d]; split-K over S -----
// Per wave: 16(m) x 64(d) tile; 4 S-chunks; 1024 waves -> 128 blocks.
__global__ void pf_gemm2_kernel(const float* __restrict__ kmat,
                                const float* __restrict__ xn,
                                float* __restrict__ ctx) {
  const int warp = threadIdx.x >> 5;
  const int l    = threadIdx.x & 31;
  const int wg   = blockIdx.x * 8 + warp;        // [0, 1024)
  const int b    = wg >> 8;
  const int rem  = wg & 255;
  const int chunk = rem >> 6;                    // 4 chunks of S
  const int rem2  = rem & 63;
  const int m0    = (rem2 >> 2) << 4;            // 16 m-tiles
  const int d0    = (rem2 & 3) << 6;             // 4 d-groups of 64
  const int lc    = l & 15;
  const int kb    = (l >> 4) << 1;
  const int sBeg  = chunk * (SS / 4);

  v8f a0 = {}, a1 = {}, a2 = {}, a3 = {};
  for (int ss = 0; ss < SS / 4; ss += 4) {
    const int s = sBeg + ss + kb;
    const size_t r0 = (size_t)(b * SS) + s;
    v2f a;
    a.x = kmat[r0 * MM + m0 + lc];
    a.y = kmat[(r0 + 1) * MM + m0 + lc];
    const float* x0 = xn + r0 * DD + d0 + lc;
    const float* x1 = x0 + DD;
    v2f b0; b0.x = x0[ 0]; b0.y = x1[ 0];
    v2f b1; b1.x = x0[16]; b1.y = x1[16];
    v2f b2; b2.x = x0[32]; b2.y = x1[32];
    v2f b3; b3.x = x0[48]; b3.y = x1[48];
    a0 = wmma4(a, b0, a0);
    a1 = wmma4(a, b1, a1);
    a2 = wmma4(a, b2, a2);
    a3 = wmma4(a, b3, a3);
  }
  const int rhalf = (l >> 4) << 3;
  for (int r = 0; r < 8; ++r) {
    size_t o = (size_t)(b * MM + m0 + r + rhalf) * DD + d0 + lc;
    atomicAdd(&ctx[o +  0], a0[r]);
    atomicAdd(&ctx[o + 16], a1[r]);
    atomicAdd(&ctx[o + 32], a2[r]);
    atomicAdd(&ctx[o + 48], a3[r]);
  }
}

// ---------------- d_inv[row] = 1 / sum_m q[row,m]*ksum[b,m] ----------------
__global__ void pf_dinv_kernel(const float* __restrict__ q,
                               const float* __restrict__ ksum,
                               float* __restrict__ dinv) {
  __shared__ float sd[256];
  const int row = blockIdx.x;
  const int t   = threadIdx.x;
  const int b   = row >> 12;                     // row / 4096
  sd[t] = q[(size_t)row * MM + t] * ksum[b * MM + t];
  __syncthreads();
  for (int s = 128; s > 0; s >>= 1) { if (t < s) sd[t] += sd[t + s]; __syncthreads(); }
  if (t == 0) dinv[row] = 1.0f / sd[0];
}

// ---------------- GEMM3: out[s,d] = dinv[s] * sum_m q[s,m]*ctx[m,d] ----------
// Written to all H head slots with non-temporal stores (write-once stream;
// keeps the ~49 MiB xn/P/q/ctx working set resident in the 192 MB L2).
__global__ void pf_gemm3_kernel(const float* __restrict__ q,
                                const float* __restrict__ ctx,
                                const float* __restrict__ dinv,
                                float* __restrict__ out) {
  const int warp = threadIdx.x >> 5;
  const int l    = threadIdx.x & 31;
  const int wg   = blockIdx.x * 8 + warp;        // [0, 4096)
  const int b    = wg >> 10;
  const int rem  = wg & 1023;
  const int s0   = (rem >> 2) << 4;
  const int d0   = (rem & 3) << 6;
  const int lc   = l & 15;
  const int kb   = (l >> 4) << 1;

  const float* Abase = q + ((size_t)(b * SS) + s0 + lc) * MM + kb;
  v8f a0 = {}, a1 = {}, a2 = {}, a3 = {};
  for (int k0 = 0; k0 < MM; k0 += 4) {
    v2f a = *(const v2f*)(Abase + k0);
    const float* c0 = ctx + (size_t)(b * MM + k0 + kb) * DD + d0 + lc;
    const float* c1 = c0 + DD;
    v2f b0; b0.x = c0[ 0]; b0.y = c1[ 0];
    v2f b1; b1.x = c0[16]; b1.y = c1[16];
    v2f b2; b2.x = c0[32]; b2.y = c1[32];
    v2f b3; b3.x = c0[48]; b3.y = c1[48];
    a0 = wmma4(a, b0, a0);
    a1 = wmma4(a, b1, a1);
    a2 = wmma4(a, b2, a2);
    a3 = wmma4(a, b3, a3);
  }
  const int rhalf = (l >> 4) << 3;
  for (int r = 0; r < 8; ++r) {
    const int srow = s0 + r + rhalf;
    const float di = dinv[b * SS + srow];
    const float v0 = a0[r] * di, v1 = a1[r] * di, v2 = a2[r] * di, v3 = a3[r] * di;
    for (int h = 0; h < HH; ++h) {
      size_t o = (((size_t)(b * HH + h) * SS) + srow) * DD + d0 + lc;
      __builtin_nontemporal_store(v0, &out[o +  0]);
      __builtin_nontemporal_store(v1, &out[o + 16]);
      __builtin_nontemporal_store(v2, &out[o + 32]);
      __builtin_nontemporal_store(v3, &out[o + 48]);
    }
  }
}

extern "C" void kernel_launch(void* const* d_in, const int* in_sizes, int n_in,
                              void* d_out, int out_size, void* d_ws, size_t ws_size,
                              hipStream_t stream) {
  const float* x     = (const float*)d_in[0];
  const float* gamma = (const float*)d_in[1];
  const float* beta  = (const float*)d_in[2];
  const float* proj  = (const float*)d_in[3];
  float* out = (float*)d_out;

  float* ws     = (float*)d_ws;
  float* xn     = ws;                               // B*S*D
  float* P      = xn  + (size_t)BB * SS * DD;       // B*S*M  (k written in-place)
  float* q      = P   + (size_t)BB * SS * MM;       // B*S*M
  float* ctx    = q   + (size_t)BB * SS * MM;       // B*M*D
  float* rowmax = ctx + (size_t)BB * MM * DD;       // B*S
  float* diag   = rowmax + BB * SS;                 // B*S
  float* dinv   = diag   + BB * SS;                 // B*S
  float* ksum   = dinv   + BB * SS;                 // B*M
  float* mx     = ksum   + BB * MM;                 // B

  pf_ln_kernel   <<<BB * SS, 256, 0, stream>>>(x, gamma, beta, xn, diag);
  pf_gemm1_kernel<<<512, 256, 0, stream>>>(xn, proj, P);
  pf_qrow_kernel <<<BB * SS, 256, 0, stream>>>(P, diag, q, rowmax);
  pf_bmax_kernel <<<BB, 256, 0, stream>>>(rowmax, mx);
  pf_zero_kernel <<<(BB * MM + 255) / 256, 256, 0, stream>>>(ksum, BB * MM);
  pf_zero_kernel <<<(BB * MM * DD + 255) / 256, 256, 0, stream>>>(ctx, BB * MM * DD);
  pf_kexp_kernel <<<BB * 16, 256, 0, stream>>>(P, diag, mx, ksum);
  pf_gemm2_kernel<<<128, 256, 0, stream>>>(P, xn, ctx);
  pf_dinv_kernel <<<BB * SS, 256, 0, stream>>>(q, ksum, dinv);
  pf_gemm3_kernel<<<512, 256, 0, stream>>>(q, ctx, dinv, out);
}